// ConstrainedLoss_1185410974162
// MI455X (gfx1250) — compile-verified
//
#include <hip/hip_runtime.h>
#include <cstdint>

#define LENGTH   128
#define NBATCH   256
#define NCELLS   8256        // 128*129/2
#define TOTAL_S  349504      // sum_{l=1..127} (128-l)*l
#define NCONSTR  8
#define BONUS_F  1000.0f
#define MAXSLAB  4096        // max (128-l)*l at l=64
#define SLABF    (MAXSLAB + 4)          // +4 floats for alignment pad
#define CHART_BYTES (NCELLS * 8)
#define SMEM_BYTES  (CHART_BYTES + 2 * SLABF * 4)

#define GLOBAL_AS __attribute__((address_space(1)))
#define LOCAL_AS  __attribute__((address_space(3)))

typedef __attribute__((__vector_size__(4 * sizeof(int)))) int v4i;

// OFF[n] = 128n - n(n-1)/2 = n*(257-n)/2
__device__ __forceinline__ int off_of(int n) { return (n * (257 - n)) >> 1; }

__device__ __forceinline__ void async_cp_b32(const float* g, float* l) {
#if __has_builtin(__builtin_amdgcn_global_load_async_to_lds_b32)
    __builtin_amdgcn_global_load_async_to_lds_b32(
        (GLOBAL_AS int*)(g),
        (LOCAL_AS int*)(unsigned)(uintptr_t)(l), 0, 0);
#else
    unsigned lo = (unsigned)(uintptr_t)l;
    unsigned long long ga = (unsigned long long)(uintptr_t)g;
    asm volatile("global_load_async_to_lds_b32 %0, %1, off" :: "v"(lo), "v"(ga) : "memory");
#endif
}

__device__ __forceinline__ void async_cp_b128(const float* g, float* l) {
#if __has_builtin(__builtin_amdgcn_global_load_async_to_lds_b128)
    __builtin_amdgcn_global_load_async_to_lds_b128(
        (GLOBAL_AS v4i*)(g),
        (LOCAL_AS v4i*)(unsigned)(uintptr_t)(l), 0, 0);
#else
    unsigned lo = (unsigned)(uintptr_t)l;
    unsigned long long ga = (unsigned long long)(uintptr_t)g;
    asm volatile("global_load_async_to_lds_b128 %0, %1, off" :: "v"(lo), "v"(ga) : "memory");
#endif
}

// Stage S floats from gsrc[0..S) into buf[pad .. pad+S).  Caller guarantees:
// buf is 16B-aligned and (gsrc byte address) % 16 == pad*4, so global index
// i and LDS index pad+i hit 16B alignment at the same i.  b128 for the
// aligned middle, b32 head/tail (<=3 elements each).
__device__ __forceinline__ void stage_slab(const float* gsrc, float* buf,
                                           int S, int pad, int tid) {
    const int i0 = (4 - pad) & 3;                 // first 16B-aligned element
    if (tid < i0 && tid < S) async_cp_b32(gsrc + tid, buf + pad + tid);
    const int nv = (S > i0) ? ((S - i0) >> 2) : 0; // 16B chunks
    const float* gv = gsrc + i0;
    float*       lv = buf + pad + i0;
    for (int c = tid; c < nv; c += 256)
        async_cp_b128(gv + 4 * c, lv + 4 * c);
    const int done = i0 + (nv << 2);
    const int rem  = S - done;                    // may be negative if S < i0
    if (tid < rem) async_cp_b32(gsrc + done + tid, buf + pad + done + tid);
}

__device__ __forceinline__ void wait_async0() {
#if __has_builtin(__builtin_amdgcn_s_wait_asynccnt)
    __builtin_amdgcn_s_wait_asynccnt(0);
#else
    asm volatile("s_wait_asynccnt 0" ::: "memory");
#endif
}

extern "C" __global__ void __launch_bounds__(256)
cky_dual_kernel(const float* __restrict__ scores,
                const int*   __restrict__ cpos,
                float*       __restrict__ ws) {
    extern __shared__ char smem[];
    float2* chart = (float2*)smem;                  // .x = constrained, .y = plain
    float*  sbuf0 = (float*)(smem + CHART_BYTES);
    float*  sbuf1 = sbuf0 + SLABF;

    const int tid = threadIdx.x;
    const int b   = blockIdx.x;
    const float* sb = scores + (long long)b * TOTAL_S;

    // Prologue: async-stage level-1 slab (soff=0 -> pad=0) into sbuf1
    stage_slab(sb, sbuf1, LENGTH - 1, 0, tid);

    // Zero both charts
    for (int i = tid; i < NCELLS; i += 256) chart[i] = make_float2(0.f, 0.f);
    __syncthreads();
    // Seed constraints (set-semantics; duplicates write the same value)
    if (tid < NCONSTR) {
        int p = cpos[b * NCONSTR + tid];
        chart[p].x = BONUS_F;
    }
    // first loop barrier (below) makes these visible before any read

    int soff = 0;
    for (int level = 1; level < LENGTH; ++level) {
        wait_async0();          // this wave's copies for current slab done
        __syncthreads();        // all waves' copies + prev-level chart writes visible

        const int L = LENGTH - level;
        const int soff_next = soff + L * level;

        // Stage next level's slab into the other buffer (consumed last level,
        // protected by the barrier above).
        if (level + 1 < LENGTH) {
            const int Sn = (LENGTH - (level + 1)) * (level + 1);
            float* nb = ((level + 1) & 1) ? sbuf1 : sbuf0;
            stage_slab(sb + soff_next, nb, Sn, soff_next & 3, tid);
        }

        const float* xs = ((level & 1) ? sbuf1 : sbuf0) + (soff & 3);

        // Lane-group geometry: G consecutive lanes cooperate on one position.
        int G = 32;
        if (level < 32) { G = 1; while (G < level) G <<= 1; }
        const int gsh    = 31 - __clz(G);
        const int lane   = tid & (G - 1);
        const int gid    = tid >> gsh;
        const int groups = 256 >> gsh;
        const int offL   = off_of(level);

        for (int pos = gid; pos < L; pos += groups) {
            float ac = -3.0e38f, ap = -3.0e38f;
            const float* xrow = xs + pos * level;
            for (int n = lane; n < level; n += G) {
                const int li = off_of(n) + pos;
                const int ri = off_of(level - 1 - n) + pos + n + 1;
                float2 lv = chart[li];          // one b64 load -> both charts
                float2 rv = chart[ri];
                float  x  = xrow[n];
                ac = fmaxf(ac, lv.x + rv.x + x);
                ap = fmaxf(ap, lv.y + rv.y + x);
            }
            for (int o = G >> 1; o > 0; o >>= 1) {
                ac = fmaxf(ac, __shfl_xor(ac, o, 32));
                ap = fmaxf(ap, __shfl_xor(ap, o, 32));
            }
            if (lane == 0) {   // unique pos per group -> race-free, deterministic
                float2 c = chart[offL + pos];
                chart[offL + pos] = make_float2(c.x + ac, c.y + ap);
            }
        }
        soff = soff_next;
        // next iteration's barrier orders chart writes / buffer reuse
    }
    __syncthreads();
    if (tid == 0) {
        float2 top = chart[NCELLS - 1];
        ws[2 * b]     = top.x - BONUS_F * NCONSTR;  // constrained score
        ws[2 * b + 1] = top.y;                      // plain (pred) score
    }
}

extern "C" __global__ void __launch_bounds__(256)
finalize_kernel(const float* __restrict__ ws, float* __restrict__ out) {
    __shared__ float sh[NBATCH];
    __shared__ float sm[NBATCH];
    const int t = threadIdx.x;
    const float constr = ws[2 * t];
    const float pred   = ws[2 * t + 1];
    const float d      = pred - constr;
    const float mask   = (fabsf(d) >= 0.001f) ? 1.0f : 0.0f;
    const float hinge  = fmaxf(1.0f + d, 0.0f) * mask;   // MARGIN = 1.0
    sh[t] = hinge;
    sm[t] = mask;
    __syncthreads();
    for (int o = NBATCH / 2; o > 0; o >>= 1) {
        if (t < o) { sh[t] += sh[t + o]; sm[t] += sm[t + o]; }
        __syncthreads();
    }
    if (t == 0) {
        float msum = sm[0], hsum = sh[0];
        out[0] = (msum > 0.1f) ? (hsum / fmaxf(msum, 1.0f)) : hsum;
    }
}

extern "C" void kernel_launch(void* const* d_in, const int* in_sizes, int n_in,
                              void* d_out, int out_size, void* d_ws, size_t ws_size,
                              hipStream_t stream) {
    const float* scores = (const float*)d_in[0];
    const int*   cpos   = (const int*)d_in[1];
    float* ws  = (float*)d_ws;
    float* out = (float*)d_out;

    (void)hipFuncSetAttribute(reinterpret_cast<const void*>(cky_dual_kernel),
                              hipFuncAttributeMaxDynamicSharedMemorySize, SMEM_BYTES);

    cky_dual_kernel<<<NBATCH, 256, SMEM_BYTES, stream>>>(scores, cpos, ws);
    finalize_kernel<<<1, 256, 0, stream>>>(ws, out);
}